// GNN_SP_49134425866247
// MI455X (gfx1250) — compile-verified
//
#include <hip/hip_runtime.h>
#include <math.h>

#define Dd 128
#define Hh 128
#define Oo 64

typedef __attribute__((ext_vector_type(2))) float v2f;
typedef __attribute__((ext_vector_type(8))) float v8f;

// ---------------------------------------------------------------- zero
__global__ void zero_kernel(float* __restrict__ p, int n) {
    int i = blockIdx.x * 256 + threadIdx.x;
    if (i < n) p[i] = 0.0f;
}

// ---------------------------------------------------------------- scatter-add (mean numerator + degree)
// one thread per (edge, feature); 128 features per edge
__global__ void scatter_kernel(const float* __restrict__ feat,
                               const int* __restrict__ src,
                               const int* __restrict__ dst,
                               float* __restrict__ agg,
                               float* __restrict__ cnt,
                               int nE) {
    int tid = blockIdx.x * 256 + threadIdx.x;
    int e = tid >> 7;
    if (e >= nE) return;
    int f = tid & 127;
    int s = src[e];
    int d = dst[e];
    unsafeAtomicAdd(&agg[(size_t)d * Hh + f], feat[(size_t)s * Hh + f]);
    if (f == 0) unsafeAtomicAdd(&cnt[d], 1.0f);
}

// Stage a weight matrix [Kdim x Ncols] into LDS with K-pair interleave:
// sWp[p*Ncols + n] = ( W[2p][n], W[2p+1][n] )
// so the WMMA B-operand is one aligned ds_load_b64.
template <int Kdim, int Ncols>
__device__ inline void stage_pairs(v2f* __restrict__ sWp, const float* __restrict__ W, int tid) {
    constexpr int total = (Kdim / 2) * Ncols;
    for (int idx = tid; idx < total; idx += 256) {
        int p = idx / Ncols;
        int n = idx - p * Ncols;
        v2f w;
        w.x = W[(2 * p) * Ncols + n];
        w.y = W[(2 * p + 1) * Ncols + n];
        sWp[idx] = w;
    }
}

// ---------------------------------------------------------------- SAGE layer:
// out = relu( L2norm( (agg/cnt) @ Wl + bl + x @ Wr ) ), H = 128
// 256 threads = 8 waves; each wave computes a 16-row x 128-col slab with
// 8 WMMA_F32_16X16X4_F32 accumulators, K-loop of 32 steps per weight matrix.
__global__ void __launch_bounds__(256)
sage_kernel(const float* __restrict__ xin,
            const float* __restrict__ agg,
            const float* __restrict__ cnt,
            const float* __restrict__ Wl,
            const float* __restrict__ bl,
            const float* __restrict__ Wr,
            float* __restrict__ outp,
            int ntiles) {
    __shared__ v2f sWp[(Hh / 2) * Hh];   // 64 KB: one pair-packed weight matrix
    const int tid  = threadIdx.x;
    const int wave = tid >> 5;
    const int lane = tid & 31;
    const int m    = lane & 15;   // row within A-tile / col within B-tile
    const int h    = lane >> 4;   // half selector (K pair 0/1 vs 2/3; out rows +8)
    const int tile = blockIdx.x * 8 + wave;
    const bool active = tile < ntiles;

    // ---- stage Wl (pair-interleaved)
    stage_pairs<Hh, Hh>(sWp, Wl, tid);
    __syncthreads();

    v8f acc[8] = {};
    float rowscale = 1.0f;
    const float* aggRow = nullptr;
    const float* xRow = nullptr;
    if (active) {
        int row = tile * 16 + m;
        rowscale = 1.0f / fmaxf(cnt[row], 1.0f);
        aggRow = agg + (size_t)row * Hh;
        xRow   = xin + (size_t)row * Hh;
    }

    // ---- phase 1: (agg/cnt) @ Wl
    if (active) {
#pragma unroll 2
        for (int k = 0; k < Hh; k += 4) {
            v2f a = *(const v2f*)(aggRow + k + 2 * h);
            a.x *= rowscale;
            a.y *= rowscale;
            const v2f* bp = &sWp[(k / 2 + h) * Hh + m];
#pragma unroll
            for (int t = 0; t < 8; ++t) {
                acc[t] = __builtin_amdgcn_wmma_f32_16x16x4_f32(
                    false, a, false, bp[t * 16], (short)0, acc[t], false, false);
            }
        }
    }
    __syncthreads();

    // ---- stage Wr (pair-interleaved)
    stage_pairs<Hh, Hh>(sWp, Wr, tid);
    __syncthreads();

    // ---- phase 2: + x @ Wr, then epilogue
    if (active) {
#pragma unroll 2
        for (int k = 0; k < Hh; k += 4) {
            v2f a = *(const v2f*)(xRow + k + 2 * h);
            const v2f* bp = &sWp[(k / 2 + h) * Hh + m];
#pragma unroll
            for (int t = 0; t < 8; ++t) {
                acc[t] = __builtin_amdgcn_wmma_f32_16x16x4_f32(
                    false, a, false, bp[t * 16], (short)0, acc[t], false, false);
            }
        }

        // bias (per output column t*16+m)
#pragma unroll
        for (int t = 0; t < 8; ++t) {
            float bv = bl[t * 16 + m];
#pragma unroll
            for (int r = 0; r < 8; ++r) acc[t][r] += bv;
        }

        // L2 normalize each row (row = tile*16 + r + 8*h), then ReLU.
        // Reduce squares across the 16 lanes of each half via xor shuffles.
#pragma unroll
        for (int r = 0; r < 8; ++r) {
            float s = 0.0f;
#pragma unroll
            for (int t = 0; t < 8; ++t) s += acc[t][r] * acc[t][r];
            s += __shfl_xor(s, 1);
            s += __shfl_xor(s, 2);
            s += __shfl_xor(s, 4);
            s += __shfl_xor(s, 8);
            float inv = 1.0f / fmaxf(sqrtf(s), 1e-12f);
#pragma unroll
            for (int t = 0; t < 8; ++t) {
                float v = acc[t][r] * inv;
                acc[t][r] = v > 0.0f ? v : 0.0f;
            }
        }

        // store: element (row = tile*16 + r + 8h, col = t*16 + m)
#pragma unroll
        for (int r = 0; r < 8; ++r) {
            int orow = tile * 16 + r + 8 * h;
            float* op = outp + (size_t)orow * Hh + m;
#pragma unroll
            for (int t = 0; t < 8; ++t) op[t * 16] = acc[t][r];
        }
    }
}

// ---------------------------------------------------------------- head:
// out = (agg/cnt) @ linW + linb   (128 -> 64), no norm/relu
__global__ void __launch_bounds__(256)
head_kernel(const float* __restrict__ agg,
            const float* __restrict__ cnt,
            const float* __restrict__ W,
            const float* __restrict__ bias,
            float* __restrict__ outp,
            int ntiles) {
    __shared__ v2f sWp[(Hh / 2) * Oo];   // 32 KB pair-packed
    const int tid  = threadIdx.x;
    const int wave = tid >> 5;
    const int lane = tid & 31;
    const int m    = lane & 15;
    const int h    = lane >> 4;
    const int tile = blockIdx.x * 8 + wave;
    const bool active = tile < ntiles;

    stage_pairs<Hh, Oo>(sWp, W, tid);
    __syncthreads();

    if (active) {
        int row = tile * 16 + m;
        float rowscale = 1.0f / fmaxf(cnt[row], 1.0f);
        const float* aRow = agg + (size_t)row * Hh;

        v8f acc[4] = {};
#pragma unroll 2
        for (int k = 0; k < Hh; k += 4) {
            v2f a = *(const v2f*)(aRow + k + 2 * h);
            a.x *= rowscale;
            a.y *= rowscale;
            const v2f* bp = &sWp[(k / 2 + h) * Oo + m];
#pragma unroll
            for (int t = 0; t < 4; ++t) {
                acc[t] = __builtin_amdgcn_wmma_f32_16x16x4_f32(
                    false, a, false, bp[t * 16], (short)0, acc[t], false, false);
            }
        }

#pragma unroll
        for (int t = 0; t < 4; ++t) {
            float bv = bias[t * 16 + m];
#pragma unroll
            for (int r = 0; r < 8; ++r) acc[t][r] += bv;
        }

#pragma unroll
        for (int r = 0; r < 8; ++r) {
            int orow = tile * 16 + r + 8 * h;
            float* op = outp + (size_t)orow * Oo + m;
#pragma unroll
            for (int t = 0; t < 4; ++t) op[t * 16] = acc[t][r];
        }
    }
}

// ---------------------------------------------------------------- launch
extern "C" void kernel_launch(void* const* d_in, const int* in_sizes, int n_in,
                              void* d_out, int out_size, void* d_ws, size_t ws_size,
                              hipStream_t stream) {
    const float* x    = (const float*)d_in[0];
    const int*   ei   = (const int*)d_in[1];   // [2,E]: row 0 src, row 1 dst
    const int*   kei  = (const int*)d_in[2];   // [2,K]
    const float* Wl1  = (const float*)d_in[3];
    const float* bl1  = (const float*)d_in[4];
    const float* Wr1  = (const float*)d_in[5];
    const float* Wl2  = (const float*)d_in[6];
    const float* bl2  = (const float*)d_in[7];
    const float* Wr2  = (const float*)d_in[8];
    const float* linW = (const float*)d_in[9];
    const float* linb = (const float*)d_in[10];
    float* out = (float*)d_out;

    const int N = in_sizes[0] / Dd;
    const int E = in_sizes[1] / 2;
    const int K = in_sizes[2] / 2;

    float* agg = (float*)d_ws;                 // N*H
    float* cnt = agg + (size_t)N * Hh;         // N   (adjacent -> zero together)
    float* z1  = cnt + N;                      // N*H
    float* z2  = z1 + (size_t)N * Hh;          // N*H

    const int nZero   = N * Hh + N;
    const int gZero   = (nZero + 255) / 256;
    const int gScatE  = (int)(((long long)E * Hh + 255) / 256);
    const int gScatK  = (int)(((long long)K * Hh + 255) / 256);
    const int ntiles  = (N + 15) / 16;         // N=50000 -> 3125 exact
    const int gGemm   = (ntiles + 7) / 8;

    // ---- layer 1
    zero_kernel<<<gZero, 256, 0, stream>>>(agg, nZero);
    scatter_kernel<<<gScatE, 256, 0, stream>>>(x, ei, ei + E, agg, cnt, E);
    sage_kernel<<<gGemm, 256, 0, stream>>>(x, agg, cnt, Wl1, bl1, Wr1, z1, ntiles);

    // ---- layer 2
    zero_kernel<<<gZero, 256, 0, stream>>>(agg, nZero);
    scatter_kernel<<<gScatE, 256, 0, stream>>>(z1, ei, ei + E, agg, cnt, E);
    sage_kernel<<<gGemm, 256, 0, stream>>>(z1, agg, cnt, Wl2, bl2, Wr2, z2, ntiles);

    // ---- k-hop mean pooling + head
    zero_kernel<<<gZero, 256, 0, stream>>>(agg, nZero);
    scatter_kernel<<<gScatK, 256, 0, stream>>>(z2, kei, kei + K, agg, cnt, K);
    head_kernel<<<gGemm, 256, 0, stream>>>(agg, cnt, linW, linb, out, ntiles);
}